// SPlineNet_35467839931023
// MI455X (gfx1250) — compile-verified
//
#include <hip/hip_runtime.h>
#include <math.h>

#define KS      5
#define KTOT    25
#define NND     50000
#define NE      1600000
#define NGR     64
#define FIN     32
#define HH1     32
#define HH2     64
#define NCLS    10
#define KBIG    832          // 25*32 spline slots + 32 root columns
#define MTILES  (NND / 16)   // 3125, exact
#define LDK     836          // padded LDS row stride (dwords): conflict-free b64 reads

typedef __attribute__((ext_vector_type(2))) float v2f;
typedef __attribute__((ext_vector_type(8))) float v8f;

// ---------------- small utility kernels ----------------

__global__ void zero_f32(float* __restrict__ p, int n) {
    int i = blockIdx.x * blockDim.x + threadIdx.x;
    if (i < n) p[i] = 0.0f;
}

__global__ void degree_k(const int* __restrict__ dst, float* __restrict__ deg, int E) {
    int e = blockIdx.x * blockDim.x + threadIdx.x;
    if (e < E) atomicAdd(&deg[dst[e]], 1.0f);
}

// Per-edge: open B-spline basis (deg-1, dim-2, ks=5), 4 nonzero slots.
// Pre-divides by deg[dst] (fuses scatter-mean) and precomputes the flat
// float offset into T: dst*832 + kidx*32.
__global__ void basis_k(const float* __restrict__ attr, const int* __restrict__ dst,
                        const float* __restrict__ deg,
                        float* __restrict__ bs, int* __restrict__ segOff, int E) {
    int e = blockIdx.x * blockDim.x + threadIdx.x;
    if (e >= E) return;
    float v0 = attr[2 * e + 0] * (float)(KS - 1);
    float v1 = attr[2 * e + 1] * (float)(KS - 1);
    float fl0 = floorf(v0), fl1 = floorf(v1);
    float f0 = v0 - fl0, f1 = v1 - fl1;
    int l0 = (int)fl0, l1 = (int)fl1;
    int i00 = min(l0, KS - 1), i01 = min(l0 + 1, KS - 1);
    int i10 = min(l1, KS - 1), i11 = min(l1 + 1, KS - 1);
    int d = dst[e];
    float inv = 1.0f / fmaxf(deg[d], 1.0f);
    float a0 = (1.0f - f0) * inv, a1 = f0 * inv;
    float c0 = 1.0f - f1, c1 = f1;
    int base = d * KBIG;
    bs[4 * e + 0] = a0 * c0;  segOff[4 * e + 0] = base + (i00 + KS * i10) * 32;
    bs[4 * e + 1] = a0 * c1;  segOff[4 * e + 1] = base + (i00 + KS * i11) * 32;
    bs[4 * e + 2] = a1 * c0;  segOff[4 * e + 2] = base + (i01 + KS * i10) * 32;
    bs[4 * e + 3] = a1 * c1;  segOff[4 * e + 3] = base + (i01 + KS * i11) * 32;
}

// Zero the 800 spline columns of T, copy node features into the 32 root columns.
__global__ void fill_T(float* __restrict__ T, const float* __restrict__ feat) {
    int k = blockIdx.x * blockDim.x + threadIdx.x;
    int m = blockIdx.y;
    if (k >= KBIG) return;
    T[(size_t)m * KBIG + k] = (k < KTOT * 32) ? 0.0f : feat[m * 32 + (k - KTOT * 32)];
}

// One lane per (edge, slot, channel) atomic add into T.
__global__ void scatter_k(const float* __restrict__ bs, const int* __restrict__ segOff,
                          const int* __restrict__ src, const float* __restrict__ feat,
                          float* __restrict__ T) {
    int idx = blockIdx.x * blockDim.x + threadIdx.x;
    int pair = idx >> 5;
    int c = idx & 31;
    if (pair >= NE * 4) return;
    int e = pair >> 2;
    float w = bs[pair];
    int off = segOff[pair];
    int s = src[e];
    atomicAdd(&T[(size_t)off + c], w * feat[s * 32 + c]);
}

// Wbig[k, o]: k<800 -> W (flat [800,Cout]); k>=800 -> root[(k-800),o]
__global__ void build_W(const float* __restrict__ W, const float* __restrict__ root,
                        float* __restrict__ Wb, int Cout) {
    int i = blockIdx.x * blockDim.x + threadIdx.x;
    if (i >= KBIG * Cout) return;
    int k = i / Cout, o = i - k * Cout;
    Wb[i] = (k < KTOT * 32) ? W[i] : root[(k - KTOT * 32) * Cout + o];
}

// ---------------- fused GEMM+bias+ReLU via V_WMMA_F32_16X16X4_F32 ----------------
// out[M, COUT] = relu(T[M, 832] @ Wb[832, COUT] + bias).
// Block = 8 waves sharing one 16-col B slice staged (transposed, padded) in LDS;
// each wave computes one 16x16 tile, 208 wmma steps over K=832.
template <int COUT>
__global__ void gemm_wmma(const float* __restrict__ T, const float* __restrict__ Wb,
                          const float* __restrict__ bias, float* __restrict__ out) {
    constexpr int NT = COUT / 16;
    __shared__ float Bl[16 * LDK];          // Bl[c*LDK + k] = B[k][n0+c]
    int mblk = blockIdx.x / NT;
    int tN   = blockIdx.x - mblk * NT;
    int n0 = tN * 16;

    // Cooperative transpose-load of the B slice into LDS (done once per block).
    for (int i = threadIdx.x; i < 16 * KBIG; i += 256) {
        int c = i & 15, k = i >> 4;
        Bl[c * LDK + k] = Wb[(size_t)k * COUT + n0 + c];
    }
    __syncthreads();

    int wave = threadIdx.x >> 5;
    int lane = threadIdx.x & 31;
    int tM = mblk * 8 + wave;
    if (tM >= MTILES) return;               // wave-uniform: EXEC all-ones for WMMA
    int m0 = tM * 16;
    int half = lane >> 4;                   // K half-split per ISA A/B layouts
    int r = lane & 15;                      // A: row M; B/D: column N

    const float* Arow = T + (size_t)(m0 + r) * KBIG + half * 2;
    const float* Brow = &Bl[r * LDK + half * 2];   // 8B-aligned, bank-conflict-free

    v8f acc = {};
    for (int k = 0; k < KBIG; k += 16) {
#pragma unroll
        for (int u = 0; u < 4; ++u) {
            int kk = k + 4 * u;
            v2f a, b;
            a.x = Arow[kk];                 // K = kk + 2*half     (global b64)
            a.y = Arow[kk + 1];             // K = kk + 2*half + 1
            b.x = Brow[kk];                 // same K pair, col n0+r  (ds b64)
            b.y = Brow[kk + 1];
            acc = __builtin_amdgcn_wmma_f32_16x16x4_f32(
                false, a, false, b, (short)0, acc, false, false);
        }
    }

    float bb = bias[n0 + r];
    float* orow = out + (size_t)(m0 + half * 8) * COUT + n0 + r;
#pragma unroll
    for (int j = 0; j < 8; ++j)             // D: VGPR j holds rows M = j and j+8
        orow[(size_t)j * COUT] = fmaxf(acc[j] + bb, 0.0f);
}

// ---------------- pooling + MLP head ----------------

__global__ void pool_k(const float* __restrict__ h, const int* __restrict__ batch,
                       float* __restrict__ gsum, float* __restrict__ cnt) {
    int i = blockIdx.x * blockDim.x + threadIdx.x;
    if (i >= NND * HH2) return;
    int n = i >> 6, c = i & 63;
    int b = batch[n];
    atomicAdd(&gsum[b * HH2 + c], h[i]);
    if (c == 0) atomicAdd(&cnt[b], 1.0f);
}

__global__ void head_k(const float* __restrict__ gsum, const float* __restrict__ cnt,
                       const float* __restrict__ w1, const float* __restrict__ b1,
                       const float* __restrict__ w2, const float* __restrict__ b2,
                       const float* __restrict__ w3, const float* __restrict__ b3,
                       float* __restrict__ out) {
    int b = threadIdx.x;                    // one thread per graph, 64 threads
    if (b >= NGR) return;
    float inv = 1.0f / fmaxf(cnt[b], 1.0f);
    float g[HH2], t[HH2];
    for (int i = 0; i < HH2; ++i) g[i] = gsum[b * HH2 + i] * inv;
    for (int o = 0; o < HH2; ++o) {
        float s = b1[o];
        for (int i = 0; i < HH2; ++i) s += g[i] * w1[i * HH2 + o];
        t[o] = g[o] * (1.0f / (1.0f + expf(-s)));
    }
    for (int o = 0; o < HH2; ++o) {
        float s = b2[o];
        for (int i = 0; i < HH2; ++i) s += t[i] * w2[i * HH2 + o];
        g[o] = t[o] * (1.0f / (1.0f + expf(-s)));
    }
    float logits[NCLS];
    float mx = -1e30f;
    for (int c = 0; c < NCLS; ++c) {
        float s = b3[c];
        for (int i = 0; i < HH2; ++i) s += g[i] * w3[i * NCLS + c];
        logits[c] = s;
        mx = fmaxf(mx, s);
    }
    float lse = 0.0f;
    for (int c = 0; c < NCLS; ++c) lse += expf(logits[c] - mx);
    lse = logf(lse);
    for (int c = 0; c < NCLS; ++c) out[b * NCLS + c] = logits[c] - mx - lse;
}

// ---------------- driver ----------------

extern "C" void kernel_launch(void* const* d_in, const int* in_sizes, int n_in,
                              void* d_out, int out_size, void* d_ws, size_t ws_size,
                              hipStream_t stream) {
    const float* x     = (const float*)d_in[0];
    const int*   eidx  = (const int*)  d_in[1];
    const float* eattr = (const float*)d_in[2];
    const int*   batch = (const int*)  d_in[3];
    const float* W1    = (const float*)d_in[4];
    const float* root1 = (const float*)d_in[5];
    const float* b1    = (const float*)d_in[6];
    const float* W2    = (const float*)d_in[7];
    const float* root2 = (const float*)d_in[8];
    const float* b2    = (const float*)d_in[9];
    const float* l1w   = (const float*)d_in[10];
    const float* l1b   = (const float*)d_in[11];
    const float* l2w   = (const float*)d_in[12];
    const float* l2b   = (const float*)d_in[13];
    const float* l3w   = (const float*)d_in[14];
    const float* l3b   = (const float*)d_in[15];
    const int* src = eidx;
    const int* dst = eidx + NE;

    char* w = (char*)d_ws;
    float* bs   = (float*)w;  w += sizeof(float) * (size_t)NE * 4;
    int*   seg  = (int*)w;    w += sizeof(int)   * (size_t)NE * 4;
    float* deg  = (float*)w;  w += sizeof(float) * (size_t)NND;
    float* T    = (float*)w;  w += sizeof(float) * (size_t)NND * KBIG;
    float* h1   = (float*)w;  w += sizeof(float) * (size_t)NND * HH1;
    float* h2   = (float*)w;  w += sizeof(float) * (size_t)NND * HH2;
    float* Wb1  = (float*)w;  w += sizeof(float) * (size_t)KBIG * HH1;
    float* Wb2  = (float*)w;  w += sizeof(float) * (size_t)KBIG * HH2;
    float* gsum = (float*)w;  w += sizeof(float) * (size_t)NGR * HH2;
    float* cnt  = (float*)w;  w += sizeof(float) * (size_t)NGR;

    const int B = 256;

    // degree + scaled basis + segment offsets (shared by both layers)
    zero_f32<<<(NND + B - 1) / B, B, 0, stream>>>(deg, NND);
    zero_f32<<<(NGR * HH2 + NGR + B - 1) / B, B, 0, stream>>>(gsum, NGR * HH2 + NGR);
    degree_k<<<(NE + B - 1) / B, B, 0, stream>>>(dst, deg, NE);
    basis_k<<<(NE + B - 1) / B, B, 0, stream>>>(eattr, dst, deg, bs, seg, NE);

    // augmented weight matrices [832, Cout]
    build_W<<<(KBIG * HH1 + B - 1) / B, B, 0, stream>>>(W1, root1, Wb1, HH1);
    build_W<<<(KBIG * HH2 + B - 1) / B, B, 0, stream>>>(W2, root2, Wb2, HH2);

    dim3 fgrid((KBIG + B - 1) / B, NND);
    int scatterBlocks = (int)(((size_t)NE * 4 * 32 + B - 1) / B);
    int mblocks = (MTILES + 7) / 8;   // 391

    // ---- layer 1 ----
    fill_T<<<fgrid, B, 0, stream>>>(T, x);
    scatter_k<<<scatterBlocks, B, 0, stream>>>(bs, seg, src, x, T);
    gemm_wmma<HH1><<<mblocks * (HH1 / 16), B, 0, stream>>>(T, Wb1, b1, h1);

    // ---- layer 2 ----
    fill_T<<<fgrid, B, 0, stream>>>(T, h1);
    scatter_k<<<scatterBlocks, B, 0, stream>>>(bs, seg, src, h1, T);
    gemm_wmma<HH2><<<mblocks * (HH2 / 16), B, 0, stream>>>(T, Wb2, b2, h2);

    // ---- pool + head ----
    pool_k<<<(NND * HH2 + B - 1) / B, B, 0, stream>>>(h2, batch, gsum, cnt);
    head_k<<<1, NGR, 0, stream>>>(gsum, cnt, l1w, l1b, l2w, l2b, l3w, l3b, (float*)d_out);
}